// VectorQuantizer_55894704390445
// MI455X (gfx1250) — compile-verified
//
#include <hip/hip_runtime.h>

#define K_CODES 1024
#define DIM 64
#define N_ROWS 65536
#define HWSZ 4096

typedef __attribute__((ext_vector_type(2))) float v2f;
typedef __attribute__((ext_vector_type(4))) float v4f;
typedef __attribute__((ext_vector_type(8))) float v8f;

#if defined(__has_builtin)
#if __has_builtin(__builtin_amdgcn_global_load_async_to_lds_b128)
#define HAVE_ASYNC 1
#endif
#endif
#ifndef HAVE_ASYNC
#define HAVE_ASYNC 0
#endif

#if HAVE_ASYNC
typedef int v4i_t __attribute__((vector_size(16)));
typedef v4i_t __attribute__((address_space(1))) gas_v4i;  // global
typedef v4i_t __attribute__((address_space(3))) las_v4i;  // LDS
#endif

__device__ __forceinline__ void atomAdd(float* p, float v) { unsafeAtomicAdd(p, v); }

__device__ __forceinline__ void wait_async0() {
#if HAVE_ASYNC
#if __has_builtin(__builtin_amdgcn_s_wait_asynccnt)
  __builtin_amdgcn_s_wait_asynccnt(0);
#else
  asm volatile("s_wait_asynccnt 0x0" ::: "memory");
#endif
#endif
}

// ---------------------------------------------------------------------------
// 0) init: zero counts/dw/accumulators, precompute w2[k] = ||W_k||^2
// ---------------------------------------------------------------------------
__global__ __launch_bounds__(256) void k_init(const float* __restrict__ W,
                                              float* __restrict__ counts,
                                              float* __restrict__ w2,
                                              float* __restrict__ scal,
                                              float* __restrict__ dw) {
  int gid = blockIdx.x * 256 + threadIdx.x;
  dw[gid] = 0.0f;
  if (gid < K_CODES) {
    counts[gid] = 0.0f;
    const float* wr = W + gid * DIM;
    float s = 0.0f;
#pragma unroll
    for (int d = 0; d < DIM; ++d) s = fmaf(wr[d], wr[d], s);
    w2[gid] = s;
  }
  if (gid < 4) scal[gid] = 0.0f;
}

// ---------------------------------------------------------------------------
// 1) transpose [B, D, H*W] -> x[N, D] contiguous rows (LDS tiled, padded)
// ---------------------------------------------------------------------------
__global__ __launch_bounds__(256) void k_transpose(const float* __restrict__ in,
                                                   float* __restrict__ xt) {
  __shared__ float s[64][65];
  const int tid = threadIdx.x;
  const int b   = blockIdx.x >> 6;
  const int hwc = (blockIdx.x & 63) << 6;
  const int q = tid >> 6;
  const int l = tid & 63;
  const float* src = in + b * (DIM * HWSZ) + hwc;
#pragma unroll
  for (int i = 0; i < 16; ++i) {
    int d = i * 4 + q;
    s[d][l] = src[d * HWSZ + l];
  }
  __syncthreads();
  float* dst = xt + (b * HWSZ + hwc) * DIM;
#pragma unroll
  for (int i = 0; i < 16; ++i) {
    int hwo = i * 4 + q;
    dst[hwo * DIM + l] = s[l][hwo];
  }
}

// ---------------------------------------------------------------------------
// 2) WMMA f32 16x16x4 GEMM + argmin + counts/dw scatter
//    block = 256 (8 wave32s); wave owns one 16-row tile; grid = 512
//    codebook streamed through double-buffered LDS (2 x 34 KB)
// ---------------------------------------------------------------------------
#define SWPITCH 68  // padded row pitch: 16 B-frag lanes hit 16 distinct banks

__device__ __forceinline__ void stage_chunk_async(const float* __restrict__ Wc,
                                                  float* sbuf, int tid) {
#if HAVE_ASYNC
#pragma unroll
  for (int i = 0; i < 8; ++i) {
    int f = tid + 256 * i;  // float4 index 0..2047
    __builtin_amdgcn_global_load_async_to_lds_b128(
        (gas_v4i*)(Wc + f * 4),
        (las_v4i*)(sbuf + (f >> 4) * SWPITCH + (f & 15) * 4),
        0, 0);
  }
#endif
}

__global__ __launch_bounds__(256) void k_main(const float* __restrict__ xt,
                                              const float* __restrict__ W,
                                              const float* __restrict__ w2,
                                              int* __restrict__ idxg,
                                              float* __restrict__ counts,
                                              float* __restrict__ dw) {
  __shared__ float sW[2][128 * SWPITCH];
  const int tid  = threadIdx.x;
  const int wave = tid >> 5;
  const int lane = tid & 31;
  const int cc   = lane & 15;
  const int hi   = lane >> 4;
  const int rb   = (blockIdx.x * 8 + wave) * 16;

  // A fragments: 16x4 f32 layout -> lane cc holds row rb+cc, cols 4*dc+2*hi..+1
  v2f a[16];
  const float* xrow = xt + (rb + cc) * DIM + 2 * hi;
#pragma unroll
  for (int dc = 0; dc < 16; ++dc) a[dc] = *(const v2f*)(xrow + 4 * dc);

  float bestv[8];
  int   besti[8];
#pragma unroll
  for (int j = 0; j < 8; ++j) { bestv[j] = 3.402823466e38f; besti[j] = 0; }

  // prologue: stage chunk 0 into buffer 0
#if HAVE_ASYNC
  stage_chunk_async(W, &sW[0][0], tid);
#else
  {
    v4f st[8];
#pragma unroll
    for (int i = 0; i < 8; ++i) st[i] = ((const v4f*)W)[tid + 256 * i];
#pragma unroll
    for (int i = 0; i < 8; ++i) {
      int f = tid + 256 * i;
      *(v4f*)&sW[0][(f >> 4) * SWPITCH + (f & 15) * 4] = st[i];
    }
  }
#endif

#pragma unroll 1
  for (int chunk = 0; chunk < 8; ++chunk) {
    const int buf = chunk & 1;
#if HAVE_ASYNC
    wait_async0();       // my staging of this chunk has landed in LDS
    __syncthreads();     // everyone's staging done; previous compute done
    if (chunk < 7) stage_chunk_async(W + (chunk + 1) * 128 * DIM,
                                     &sW[buf ^ 1][0], tid);
#else
    __syncthreads();
    v4f st2[8];
    if (chunk < 7) {     // issue next chunk's global loads before compute
      const v4f* src = (const v4f*)(W + (chunk + 1) * 128 * DIM);
#pragma unroll
      for (int i = 0; i < 8; ++i) st2[i] = src[tid + 256 * i];
    }
#endif

#pragma unroll 1
    for (int kb = 0; kb < 8; ++kb) {
      const int code = chunk * 128 + kb * 16 + cc;
      const float* wrow = &sW[buf][(kb * 16 + cc) * SWPITCH + 2 * hi];
      // 4 independent accumulator chains (dependent WMMAs 4 issues apart)
      v8f acc0 = {}, acc1 = {}, acc2 = {}, acc3 = {};
#pragma unroll
      for (int g = 0; g < 4; ++g) {
        const float* wp = wrow + 16 * g;
        acc0 = __builtin_amdgcn_wmma_f32_16x16x4_f32(
            false, a[4 * g + 0], false, *(const v2f*)(wp + 0), (short)0, acc0,
            false, false);
        acc1 = __builtin_amdgcn_wmma_f32_16x16x4_f32(
            false, a[4 * g + 1], false, *(const v2f*)(wp + 4), (short)0, acc1,
            false, false);
        acc2 = __builtin_amdgcn_wmma_f32_16x16x4_f32(
            false, a[4 * g + 2], false, *(const v2f*)(wp + 8), (short)0, acc2,
            false, false);
        acc3 = __builtin_amdgcn_wmma_f32_16x16x4_f32(
            false, a[4 * g + 3], false, *(const v2f*)(wp + 12), (short)0, acc3,
            false, false);
      }
      const float w2v = w2[code];
#pragma unroll
      for (int j = 0; j < 8; ++j) {  // row M = j + 8*hi
        float s = (acc0[j] + acc1[j]) + (acc2[j] + acc3[j]);
        float cand = fmaf(-2.0f, s, w2v);  // dist proxy (x^2 dropped)
        if (cand < bestv[j]) { bestv[j] = cand; besti[j] = code; }
      }
    }

#if !HAVE_ASYNC
    if (chunk < 7) {     // drain staged registers into the other buffer
#pragma unroll
      for (int i = 0; i < 8; ++i) {
        int f = tid + 256 * i;
        *(v4f*)&sW[buf ^ 1][(f >> 4) * SWPITCH + (f & 15) * 4] = st2[i];
      }
    }
#endif
  }

  // argmin across the 16 lanes of each half (masks stay inside the half)
#pragma unroll
  for (int j = 0; j < 8; ++j) {
#pragma unroll
    for (int m = 1; m < 16; m <<= 1) {
      float ov = __shfl_xor(bestv[j], m, 32);
      int   oi = __shfl_xor(besti[j], m, 32);
      if (ov < bestv[j] || (ov == bestv[j] && oi < besti[j])) {
        bestv[j] = ov; besti[j] = oi;
      }
    }
  }

  if (cc == 0) {  // lanes 0 and 16
#pragma unroll
    for (int j = 0; j < 8; ++j) {
      idxg[rb + 8 * hi + j] = besti[j];
      atomAdd(&counts[besti[j]], 1.0f);
    }
  }

  // scatter-add rows of x into dw[code]; lane owns 2 columns for all 16 rows
#pragma unroll
  for (int r = 0; r < 16; ++r) {
    int code_r = __shfl(besti[r & 7], (r < 8) ? 0 : 16, 32);
    v2f xv = *(const v2f*)(xt + (rb + r) * DIM + 2 * lane);
    atomAdd(&dw[code_r * DIM + 2 * lane],     xv.x);
    atomAdd(&dw[code_r * DIM + 2 * lane + 1], xv.y);
  }
}

// ---------------------------------------------------------------------------
// 3) new_cluster_size + reductions for n and entropy (4 blocks x 256 = K)
// ---------------------------------------------------------------------------
__global__ __launch_bounds__(256) void k_ema_cs(const float* __restrict__ counts,
                                                const float* __restrict__ ema_cs,
                                                float* __restrict__ ncs_out,
                                                float* __restrict__ scal) {
  __shared__ float r1[256], r2[256];
  int tid = threadIdx.x;
  int k = blockIdx.x * 256 + tid;
  float cnt = counts[k];
  float ncs = fmaf(ema_cs[k], 0.99f, 0.01f * cnt);
  ncs_out[k] = ncs;
  float p = cnt * (1.0f / 65536.0f);
  float ent = -p * logf(p + 1e-10f);
  r1[tid] = ncs; r2[tid] = ent;
  __syncthreads();
  for (int s = 128; s > 0; s >>= 1) {
    if (tid < s) { r1[tid] += r1[tid + s]; r2[tid] += r2[tid + s]; }
    __syncthreads();
  }
  if (tid == 0) { atomAdd(&scal[0], r1[0]); atomAdd(&scal[1], r2[0]); }
}

// 4) new_ema_w (65536 elements)
__global__ __launch_bounds__(256) void k_ema_w(const float* __restrict__ ema_w,
                                               const float* __restrict__ dw,
                                               float* __restrict__ ema_w_out) {
  int i = blockIdx.x * 256 + threadIdx.x;
  ema_w_out[i] = fmaf(ema_w[i], 0.99f, 0.01f * dw[i]);
}

// 5) new_embed_w = new_ema_w / normalized (65536 elements)
__global__ __launch_bounds__(256) void k_embed(const float* __restrict__ ema_w_out,
                                               const float* __restrict__ ncs_out,
                                               const float* __restrict__ scal,
                                               float* __restrict__ embed_out) {
  int i = blockIdx.x * 256 + threadIdx.x;
  int k = i >> 6;
  float n = scal[0];
  float norm = (ncs_out[k] + 1e-5f) / (n + 1024.0f * 1e-5f) * n;
  embed_out[i] = ema_w_out[i] / norm;
}

// ---------------------------------------------------------------------------
// 6) quantize (== gather of updated codebook) + SSE wave32 reduction
// ---------------------------------------------------------------------------
__global__ __launch_bounds__(256) void k_quant(const float* __restrict__ in,
                                               const int* __restrict__ idxg,
                                               const float* __restrict__ embed,
                                               float* __restrict__ qout,
                                               float* __restrict__ scal) {
  __shared__ float wsum[8];
  int tid = threadIdx.x;
  int o = blockIdx.x * 256 + tid;          // flat [B,C,H,W] index
  int b  = o >> 18;
  int c  = (o >> 12) & 63;
  int hw = o & 4095;
  int r  = (b << 12) | hw;                 // flat row id in x[N,D]
  float q  = embed[idxg[r] * DIM + c];
  float xv = in[o];
  qout[o] = q;                             // straight-through forward == q
  float d = xv - q;
  float v = d * d;
#pragma unroll
  for (int m = 16; m >= 1; m >>= 1) v += __shfl_xor(v, m, 32);
  if ((tid & 31) == 0) wsum[tid >> 5] = v;
  __syncthreads();
  if (tid == 0) {
    float s = 0.0f;
#pragma unroll
    for (int i = 0; i < 8; ++i) s += wsum[i];
    atomAdd(&scal[2], s);
  }
}

// 7) finalize scalars
__global__ void k_final(const float* __restrict__ scal,
                        float* __restrict__ loss, float* __restrict__ perp) {
  *loss = 0.25f * scal[2] * (1.0f / 4194304.0f);
  *perp = expf(scal[1]);
}

// ---------------------------------------------------------------------------
extern "C" void kernel_launch(void* const* d_in, const int* in_sizes, int n_in,
                              void* d_out, int out_size, void* d_ws, size_t ws_size,
                              hipStream_t stream) {
  const float* inputs  = (const float*)d_in[0];  // [16,64,64,64]
  const float* embed_w = (const float*)d_in[1];  // [1024,64]
  const float* ema_cs  = (const float*)d_in[2];  // [1024]
  const float* ema_w   = (const float*)d_in[3];  // [1024,64]

  float* out       = (float*)d_out;
  float* quant     = out;                 // 4194304
  float* loss      = out + 4194304;       // 1
  float* perp      = out + 4194305;       // 1
  float* ncs_out   = out + 4194306;       // 1024
  float* ema_w_out = out + 4195330;       // 65536
  float* embed_out = out + 4260866;       // 65536

  float* wsf    = (float*)d_ws;
  float* counts = wsf;                    // 1024
  float* w2     = wsf + 1024;             // 1024
  float* scal   = wsf + 2048;             // [n, entropy, sse, pad]
  float* dw     = wsf + 2056;             // 65536
  int*   idxg   = (int*)(wsf + 2056 + 65536);   // 65536 ints
  float* xt     = wsf + 2056 + 65536 + 65536;   // x[N,64] = 16 MB

  k_init     <<<256,   256, 0, stream>>>(embed_w, counts, w2, scal, dw);
  k_transpose<<<1024,  256, 0, stream>>>(inputs, xt);
  k_main     <<<512,   256, 0, stream>>>(xt, embed_w, w2, idxg, counts, dw);
  k_ema_cs   <<<4,     256, 0, stream>>>(counts, ema_cs, ncs_out, scal);
  k_ema_w    <<<256,   256, 0, stream>>>(ema_w, dw, ema_w_out);
  k_embed    <<<256,   256, 0, stream>>>(ema_w_out, ncs_out, scal, embed_out);
  k_quant    <<<16384, 256, 0, stream>>>(inputs, idxg, embed_out, quant, scal);
  k_final    <<<1,     1,   0, stream>>>(scal, loss, perp);
}